// GCNLayer_20358144983127
// MI455X (gfx1250) — compile-verified
//
#include <hip/hip_runtime.h>
#include <hip/hip_bf16.h>

// GCN layer: h = X @ W  (f32 GEMM via V_WMMA_F32_16X16X4_F32), then
// out[dst] += h[src] over 1.6M edges (L2-resident f32 atomics).
//
// Sizes (from reference): N=100000 nodes, IN=128, OUT=32, E=1600000.
// N is an exact multiple of 16 -> clean 16-row WMMA tiles, EXEC all-1s.

typedef __attribute__((ext_vector_type(2))) float v2f;
typedef __attribute__((ext_vector_type(8))) float v8f;

#define IN_FEATS 128
#define OUT_FEATS 32

// ---------------------------------------------------------------------------
// Kernel 1: h = X @ W using f32 WMMA 16x16x4.
// One wave computes a 16x32 tile of h: two 16x16 f32 accumulators.
// A fragment (16x4 f32, 2 VGPRs): lane L<16 holds A[M=L][K=k0..k0+1],
//   lane L>=16 holds A[M=L-16][K=k0+2..k0+3]  -> one contiguous b64 load.
// B fragment (4x16 f32, 2 VGPRs): VGPR v, lane L: K = k0 + v + (L>=16 ? 2:0),
//   N = L&15  -> two ds_load_b32 per fragment from LDS-staged W.
// ---------------------------------------------------------------------------
__global__ void __launch_bounds__(256) gcn_gemm_wmma(
    const float* __restrict__ X, const float* __restrict__ W,
    float* __restrict__ H, int n_nodes) {
  __shared__ float wlds[IN_FEATS * OUT_FEATS];  // 16 KB

  // Stage weight into LDS cooperatively (256 threads x 4 float4 = 4096 f32).
  {
    const float4* wg = (const float4*)W;
    float4* wl = (float4*)wlds;
    for (int i = threadIdx.x; i < (IN_FEATS * OUT_FEATS) / 4; i += 256)
      wl[i] = wg[i];
  }
  __syncthreads();

  const int wave = threadIdx.x >> 5;       // 8 waves per block
  const int lane = threadIdx.x & 31;
  const int tile = blockIdx.x * 8 + wave;  // 16-row tile index
  const int n_tiles = n_nodes >> 4;
  if (tile >= n_tiles) return;             // wave-uniform: EXEC stays all-1s

  const int mrow = lane & 15;
  const int col = lane & 15;
  const int khalf = (lane >> 4) << 1;      // 0 for lanes 0-15, 2 for 16-31

  const float* arow = X + (size_t)(tile * 16 + mrow) * IN_FEATS + khalf;

  v8f acc0 = {};
  v8f acc1 = {};

#pragma unroll 4
  for (int k0 = 0; k0 < IN_FEATS; k0 += 4) {
    // A fragment: 8-byte aligned contiguous pair.
    v2f a = *(const v2f*)(arow + k0);

    // B fragments for N-tiles 0 (cols 0-15) and 1 (cols 16-31).
    const int kk = k0 + khalf;
    v2f b0, b1;
    b0.x = wlds[kk * OUT_FEATS + col];
    b0.y = wlds[(kk + 1) * OUT_FEATS + col];
    b1.x = wlds[kk * OUT_FEATS + 16 + col];
    b1.y = wlds[(kk + 1) * OUT_FEATS + 16 + col];

    // 8 args: (neg_a, A, neg_b, B, c_mod, C, reuse_a, reuse_b)
    acc0 = __builtin_amdgcn_wmma_f32_16x16x4_f32(false, a, false, b0,
                                                 (short)0, acc0, false, false);
    acc1 = __builtin_amdgcn_wmma_f32_16x16x4_f32(false, a, false, b1,
                                                 (short)0, acc1, false, false);
  }

  // C/D layout: VGPR r, lanes 0-15 -> row r, lanes 16-31 -> row r+8; N = lane&15.
  const int mbase = tile * 16 + ((lane >> 4) << 3);
  float* hrow0 = H + (size_t)mbase * OUT_FEATS + col;
#pragma unroll
  for (int r = 0; r < 8; ++r) {
    hrow0[(size_t)r * OUT_FEATS] = acc0[r];
    hrow0[(size_t)r * OUT_FEATS + 16] = acc1[r];
  }
}

// ---------------------------------------------------------------------------
// Kernel 2: zero d_out (harness poisons it; scatter accumulates into it).
// ---------------------------------------------------------------------------
__global__ void __launch_bounds__(256) gcn_zero(float4* __restrict__ out,
                                                int n_vec4) {
  int i = blockIdx.x * 256 + threadIdx.x;
  if (i < n_vec4) out[i] = float4{0.f, 0.f, 0.f, 0.f};
}

// ---------------------------------------------------------------------------
// Kernel 3: out[dst[e]] += h[src[e]].  One wave per edge, lane = feature.
// h row = 128 B coalesced load; 32 global_atomic_add_f32 per edge hit L2
// (out is 12.8 MB, L2 is 192 MB).
// ---------------------------------------------------------------------------
__global__ void __launch_bounds__(256) gcn_edge_scatter(
    const float* __restrict__ H, const int* __restrict__ src,
    const int* __restrict__ dst, float* __restrict__ out, int n_edges) {
  const int edge = (int)((blockIdx.x * 256 + threadIdx.x) >> 5);
  const int lane = threadIdx.x & 31;
  if (edge >= n_edges) return;  // wave-uniform

  const int s = src[edge];
  const int d = dst[edge];
  const float v = H[(size_t)s * OUT_FEATS + lane];
  atomicAdd(out + (size_t)d * OUT_FEATS + lane, v);
}

// ---------------------------------------------------------------------------
// Launcher. Inputs: [0] X (N*128 f32), [1] W (128*32 f32), [2] src (E i32),
// [3] dst (E i32). Output: out (N*32 f32). d_ws holds h (N*32 f32 = 12.8 MB).
// ---------------------------------------------------------------------------
extern "C" void kernel_launch(void* const* d_in, const int* in_sizes, int n_in,
                              void* d_out, int out_size, void* d_ws,
                              size_t ws_size, hipStream_t stream) {
  const float* X = (const float*)d_in[0];
  const float* W = (const float*)d_in[1];
  const int* src = (const int*)d_in[2];
  const int* dst = (const int*)d_in[3];
  float* out = (float*)d_out;

  const int n_nodes = in_sizes[0] / IN_FEATS;
  const int n_edges = in_sizes[2];

  float* H = (float*)d_ws;  // needs n_nodes*OUT_FEATS*4 bytes (12.8 MB)

  // 1) GEMM: 8 tiles (128 rows) per 256-thread block.
  const int n_tiles = n_nodes / 16;
  const int gemm_blocks = (n_tiles + 7) / 8;
  gcn_gemm_wmma<<<gemm_blocks, 256, 0, stream>>>(X, W, H, n_nodes);

  // 2) Zero the output accumulator.
  const int n_vec4 = out_size / 4;
  gcn_zero<<<(n_vec4 + 255) / 256, 256, 0, stream>>>((float4*)out, n_vec4);

  // 3) Edge scatter-add: one wave (32 lanes) per edge.
  const long long threads = (long long)n_edges * 32;
  const int scat_blocks = (int)((threads + 255) / 256);
  gcn_edge_scatter<<<scat_blocks, 256, 0, stream>>>(H, src, dst, out, n_edges);
}